// MultiHeadAttentionEdge_47442208751682
// MI455X (gfx1250) — compile-verified
//
#include <hip/hip_runtime.h>
#include <stdint.h>

// Problem constants (fixed by setup_inputs)
#define B  2
#define N  512
#define M  512
#define DQ 128
#define DE 64
#define H  8
#define O  32
#define OUT 128
#define QSCALE 0.17677669529663687f   // 1/sqrt(32)

typedef __attribute__((ext_vector_type(2))) float v2f;
typedef __attribute__((ext_vector_type(8))) float v8f;

// ---------------------------------------------------------------------------
// Kernel 1: per (b,n) compute q = (query @ Wq) * 1/sqrt(32)  -> qh [BN][H][O]
//           and qWe[h][j] = sum_o q[h][o] * Wk[h][128+j][o]  -> qwe [BN][H][DE]
// ---------------------------------------------------------------------------
__global__ __launch_bounds__(256) void k_qproj(
    const float* __restrict__ query, const float* __restrict__ Wq,
    const float* __restrict__ Wk, float* __restrict__ qh, float* __restrict__ qwe) {
  int bn = blockIdx.x;
  int t = threadIdx.x;
  __shared__ __attribute__((aligned(16))) float s_q[DQ];
  __shared__ __attribute__((aligned(16))) float s_qh[H * O];

  if (t < DQ) s_q[t] = query[(long)bn * DQ + t];
  __syncthreads();

  int h = t >> 5, o = t & 31;
  const float* wp = Wq + (long)h * DQ * O + o;
  float s = 0.f;
#pragma unroll 4
  for (int i = 0; i < DQ; ++i) s = fmaf(s_q[i], wp[(long)i * O], s);
  s *= QSCALE;
  qh[(long)bn * (H * O) + t] = s;
  s_qh[t] = s;
  __syncthreads();

#pragma unroll
  for (int rep = 0; rep < 2; ++rep) {
    int idx = t + rep * 256;          // 0..511 = h*64 + j
    int hh = idx >> 6, j = idx & 63;
    const float* wk = Wk + ((long)hh * (DQ + DE) + DQ + j) * O;
    float acc = 0.f;
#pragma unroll
    for (int o2 = 0; o2 < O; o2 += 4) {
      float4 kv = *(const float4*)(wk + o2);
      acc = fmaf(s_qh[hh * O + o2 + 0], kv.x, acc);
      acc = fmaf(s_qh[hh * O + o2 + 1], kv.y, acc);
      acc = fmaf(s_qh[hh * O + o2 + 2], kv.z, acc);
      acc = fmaf(s_qh[hh * O + o2 + 3], kv.w, acc);
    }
    qwe[(long)bn * (H * DE) + idx] = acc;
  }
}

// ---------------------------------------------------------------------------
// Kernel 2: per (b,m) compute K' = key @ Wk[:,0:128,:] and V' = value @ Wv
//           -> kkh [BM][H][O], vvh [BM][H][O]
// ---------------------------------------------------------------------------
__global__ __launch_bounds__(256) void k_kvproj(
    const float* __restrict__ key, const float* __restrict__ value,
    const float* __restrict__ Wk, const float* __restrict__ Wv,
    float* __restrict__ kkh, float* __restrict__ vvh) {
  int bm = blockIdx.x;
  int t = threadIdx.x;
  __shared__ __attribute__((aligned(16))) float s_k[DQ];
  __shared__ __attribute__((aligned(16))) float s_v[DQ];
  if (t < DQ) s_k[t] = key[(long)bm * DQ + t];
  else        s_v[t - DQ] = value[(long)bm * DQ + (t - DQ)];
  __syncthreads();

  int h = t >> 5, o = t & 31;
  const float* wk = Wk + (long)h * (DQ + DE) * O + o;   // rows 0..127 = key part
  const float* wv = Wv + (long)h * DQ * O + o;
  float sk = 0.f, sv = 0.f;
#pragma unroll 4
  for (int i = 0; i < DQ; ++i) {
    sk = fmaf(s_k[i], wk[(long)i * O], sk);
    sv = fmaf(s_v[i], wv[(long)i * O], sv);
  }
  kkh[(long)bm * (H * O) + t] = sk;
  vvh[(long)bm * (H * O) + t] = sv;
}

// ---------------------------------------------------------------------------
// Kernel 3: fused attention per (b,n). 256 threads = 8 waves.
//   edge tile streamed via GLOBAL_LOAD_ASYNC_TO_LDS_B128 (ASYNCcnt),
//   logits[h][m] = WMMA_f32( edge_tile[16x64] x qWe^T[64x16] ) + q.K' base,
//   softmax over m, then attn x V', then projection.
// ---------------------------------------------------------------------------
__global__ __launch_bounds__(256) void k_attn(
    const float* __restrict__ edge, const float* __restrict__ qh,
    const float* __restrict__ qwe,  const float* __restrict__ kkh,
    const float* __restrict__ vvh,  const float* __restrict__ proj,
    const float* __restrict__ bias, float* __restrict__ out) {
  const int bn = blockIdx.x;         // b*N + n
  const int b  = bn >> 9;            // N = 512
  const int t  = threadIdx.x;
  const int w  = t >> 5;             // wave id 0..7
  const int ln = t & 31;             // lane id

  __shared__ __attribute__((aligned(16))) float s_qwe[16 * 66];     // B-matrix, padded
  __shared__ __attribute__((aligned(16))) float s_logits[H * M];    // 16 KB
  __shared__ __attribute__((aligned(16))) float s_edge[8][16 * 68]; // per-wave tile
  __shared__ __attribute__((aligned(16))) float s_mh[H * O];

  // ---- stage qWe as B matrix (rows h>=8 are zero) ----
  for (int idx = t; idx < 16 * 64; idx += 256) {
    int h = idx >> 6, j = idx & 63;
    s_qwe[h * 66 + j] = (h < 8) ? qwe[(long)bn * (H * DE) + h * DE + j] : 0.f;
  }
  __syncthreads();

  // ---- build B fragments: for K-step kk, lane ln holds B[4kk + 2*(ln/16) + {0,1}, ln&15]
  const int hcol = ln & 15;
  const int joff = (ln >> 4) << 1;
  v2f bfrag[16];
#pragma unroll
  for (int kk = 0; kk < 16; ++kk)
    bfrag[kk] = *(const v2f*)&s_qwe[hcol * 66 + (kk << 2) + joff];

  // ---- edge-term logits via V_WMMA_F32_16X16X4_F32 ----
  float* myedge = &s_edge[w][0];
  for (int tile = w; tile < 32; tile += 8) {        // uniform per wave: 4 tiles each
    int m0 = tile << 4;
    long gbase = ((long)bn * M + m0) * DE;          // contiguous 16x64 f32 tile

    // Async direct global->LDS copy (no VGPR staging). GV mode: 64-bit vaddr,
    // LDS byte offset (low 32 bits of generic shared pointer) in a VGPR.
#pragma unroll
    for (int i = 0; i < 8; ++i) {
      int dw = (ln << 2) + (i << 7);
      int row = dw >> 6, col = dw & 63;
      unsigned ldsa = (unsigned)(uintptr_t)(&myedge[row * 68 + col]);
      const float* gp = edge + gbase + dw;
      asm volatile("global_load_async_to_lds_b128 %0, %1, off"
                   :: "v"(ldsa), "v"(gp)
                   : "memory");
    }
    asm volatile("s_wait_asynccnt 0x0" ::: "memory");

    v8f acc = {0.f, 0.f, 0.f, 0.f, 0.f, 0.f, 0.f, 0.f};
#pragma unroll
    for (int kk = 0; kk < 16; ++kk) {
      // A frag: lane ln holds edge[m0 + (ln&15), 4kk + 2*(ln/16) + {0,1}]
      v2f a = *(const v2f*)&myedge[(ln & 15) * 68 + (kk << 2) + joff];
      acc = __builtin_amdgcn_wmma_f32_16x16x4_f32(
          false, a, false, bfrag[kk], (short)0, acc, false, false);
    }
    if (hcol < 8) {                                  // D[m_local, h]: VGPR r -> M=r(+8)
      int mb = m0 + ((ln >> 4) << 3);
#pragma unroll
      for (int r = 0; r < 8; ++r) s_logits[hcol * M + mb + r] = acc[r];
    }
  }
  __syncthreads();

  // ---- add base logits: wave w owns head h=w ----
  {
    const int h = w;
    float qr[O];
#pragma unroll
    for (int i = 0; i < O; i += 4)
      *(float4*)&qr[i] = *(const float4*)(qh + (long)bn * (H * O) + h * O + i);
    for (int m = ln; m < M; m += 32) {
      const float* kr = kkh + ((long)(b * M + m) * H + h) * O;
      float s = 0.f;
#pragma unroll
      for (int i = 0; i < O; i += 4) {
        float4 kv = *(const float4*)(kr + i);
        s = fmaf(qr[i + 0], kv.x, s);
        s = fmaf(qr[i + 1], kv.y, s);
        s = fmaf(qr[i + 2], kv.z, s);
        s = fmaf(qr[i + 3], kv.w, s);
      }
      s_logits[h * M + m] += s;
    }
  }
  __syncthreads();

  // ---- softmax over m for head h=w (lane owns 16 contiguous m) ----
  {
    const int h = w;
    float v[16];
    int base = h * M + ln * 16;
#pragma unroll
    for (int i = 0; i < 16; i += 4) *(float4*)&v[i] = *(const float4*)&s_logits[base + i];
    float mx = v[0];
#pragma unroll
    for (int i = 1; i < 16; ++i) mx = fmaxf(mx, v[i]);
#pragma unroll
    for (int off = 16; off > 0; off >>= 1) mx = fmaxf(mx, __shfl_xor(mx, off));
    float sum = 0.f;
#pragma unroll
    for (int i = 0; i < 16; ++i) { v[i] = __expf(v[i] - mx); sum += v[i]; }
#pragma unroll
    for (int off = 16; off > 0; off >>= 1) sum += __shfl_xor(sum, off);
    float inv = 1.0f / sum;
#pragma unroll
    for (int i = 0; i < 16; ++i) v[i] *= inv;
#pragma unroll
    for (int i = 0; i < 16; i += 4) *(float4*)&s_logits[base + i] = *(float4*)&v[i];
  }
  __syncthreads();

  // ---- attn x V': thread (h=w, o=ln); coalesced V' reads across lanes ----
  {
    const int h = w, o = ln;
    long vbase = (long)(b * M) * (H * O) + h * O + o;
    float acc = 0.f;
    for (int m = 0; m < M; m += 4) {
      float4 a4 = *(const float4*)&s_logits[h * M + m];   // wave-uniform address
      acc = fmaf(a4.x, vvh[vbase + (long)(m + 0) * (H * O)], acc);
      acc = fmaf(a4.y, vvh[vbase + (long)(m + 1) * (H * O)], acc);
      acc = fmaf(a4.z, vvh[vbase + (long)(m + 2) * (H * O)], acc);
      acc = fmaf(a4.w, vvh[vbase + (long)(m + 3) * (H * O)], acc);
    }
    s_mh[h * O + o] = acc;
  }
  __syncthreads();

  // ---- output projection: out[bn][c] = bias[c] + sum_{h,i} mh[h,i]*proj[h,i,c]
  if (t < OUT) {
    float acc = bias[t];
#pragma unroll 4
    for (int idx = 0; idx < H * O; ++idx)
      acc = fmaf(s_mh[idx], proj[(long)idx * OUT + t], acc);
    out[(long)bn * OUT + t] = acc;
  }
}

// ---------------------------------------------------------------------------
extern "C" void kernel_launch(void* const* d_in, const int* in_sizes, int n_in,
                              void* d_out, int out_size, void* d_ws, size_t ws_size,
                              hipStream_t stream) {
  (void)in_sizes; (void)n_in; (void)out_size; (void)ws_size;
  const float* query = (const float*)d_in[0];
  const float* key   = (const float*)d_in[1];
  const float* value = (const float*)d_in[2];
  const float* edge  = (const float*)d_in[3];
  const float* Wq    = (const float*)d_in[4];
  const float* Wk    = (const float*)d_in[5];
  const float* Wv    = (const float*)d_in[6];
  const float* Wp    = (const float*)d_in[7];
  const float* pb    = (const float*)d_in[8];
  float* out = (float*)d_out;

  float* ws  = (float*)d_ws;
  float* qh  = ws;                       // B*N*H*O    = 262144 floats
  float* qwe = ws + 262144;              // B*N*H*DE   = 524288 floats
  float* kkh = ws + 262144 + 524288;     // B*M*H*O    = 262144 floats
  float* vvh = ws + 262144 + 524288 + 262144;  // B*M*H*O = 262144 floats

  k_qproj<<<B * N, 256, 0, stream>>>(query, Wq, Wk, qh, qwe);
  k_kvproj<<<B * M, 256, 0, stream>>>(key, value, Wk, Wv, kkh, vvh);
  k_attn<<<B * N, 256, 0, stream>>>(edge, qh, qwe, kkh, vvh, Wp, pb, out);
}